// GenSP_82085414961602
// MI455X (gfx1250) — compile-verified
//
#include <hip/hip_runtime.h>
#include <math.h>

// Problem constants (from reference): x = (4, 9, 256, 256) f32, S=16
#define NB   4
#define NC   9
#define HH   256
#define WW   256
#define SGRID 16          // sH == sW == 16
#define SHW  256          // superpixels per image
#define HW   65536        // pixels per image
#define CSTRIDE 16        // padded centroid row: [9 ch][3 zeros][cnorm][3 zeros]

typedef __attribute__((ext_vector_type(2))) float v2f;
typedef __attribute__((ext_vector_type(8))) float v8f;

// Device-pass-only wrapper: the amdgcn WMMA builtin is not declared during the
// host pass of the HIP TU, so keep it out of the host compiler's sight.
__device__ __forceinline__ v8f wmma_dot_16x16x4(v2f a, v2f b, v8f c) {
#if defined(__AMDGCN__)
    // 8 args: (neg_a, A, neg_b, B, c_mod, C, reuse_a, reuse_b)
    return __builtin_amdgcn_wmma_f32_16x16x4_f32(false, a, false, b, (short)0, c,
                                                 false, false);
#else
    (void)a; (void)b;
    return c;   // host pass stub — never executed
#endif
}

// ---------------------------------------------------------------------------
// Kernel 0: zero-fill the dense output (256 MB) with wide stores.
// ---------------------------------------------------------------------------
__global__ __launch_bounds__(256) void zero_kernel(float4* __restrict__ out, long n4) {
    long i = (long)blockIdx.x * blockDim.x + threadIdx.x;
    long stride = (long)gridDim.x * blockDim.x;
    float4 z; z.x = 0.f; z.y = 0.f; z.z = 0.f; z.w = 0.f;
    for (; i < n4; i += stride) out[i] = z;
}

// ---------------------------------------------------------------------------
// Kernel 1: block-mean centroids. One workgroup per (b, s); 256 threads = the
// 16x16 pixels of the block. Tree reduction in LDS (deterministic).
// Centroid row layout (16 floats): [c0..c8, 0,0,0, cnorm, 0,0,0]
// ---------------------------------------------------------------------------
__global__ __launch_bounds__(256) void centroid_mean_kernel(const float* __restrict__ x,
                                                            float* __restrict__ cent) {
    int s = blockIdx.x & (SHW - 1);
    int b = blockIdx.x >> 8;
    int tr = s >> 4, tc = s & 15;
    int tid = threadIdx.x;
    int py = tid >> 4, px = tid & 15;
    int y = tr * 16 + py, xc = tc * 16 + px;

    const float* xb = x + (size_t)b * NC * HW;
    __shared__ float red[256];
    __shared__ float mean[NC];

    #pragma unroll
    for (int c = 0; c < NC; ++c) {
        red[tid] = xb[(size_t)c * HW + y * WW + xc];
        __syncthreads();
        for (int off = 128; off > 0; off >>= 1) {
            if (tid < off) red[tid] += red[tid + off];
            __syncthreads();
        }
        if (tid == 0) mean[c] = red[0] * (1.0f / 256.0f);
        __syncthreads();
    }
    if (tid == 0) {
        float* row = cent + ((size_t)b * SHW + s) * CSTRIDE;
        float cn = 0.f;
        #pragma unroll
        for (int c = 0; c < NC; ++c) { row[c] = mean[c]; cn += mean[c] * mean[c]; }
        row[9] = 0.f; row[10] = 0.f; row[11] = 0.f;
        row[12] = cn;
        row[13] = 0.f; row[14] = 0.f; row[15] = 0.f;
    }
}

// ---------------------------------------------------------------------------
// Kernel 2/4: affinity via WMMA. One wave (32 threads) per 16-pixel row
// segment of a block. Computes dot(pixel, neighbor-centroid) as a
// 16x16x12 f32 GEMM = 3 chained v_wmma_f32_16x16x4_f32, then masked softmax
// over the 9 neighbors (|pf|^2 cancels in the softmax).
// All guard conditions are applied as VALU selects on loaded VALUES (the
// padded addresses are always in-bounds), so the loads stay unconditional —
// no exec-mask branching around VMEM.
//   write_dense == 0 : store aff[b][n][p]   (iteration 0)
//   write_dense == 1 : scatter into dense[b][spix][p]  (iteration 1)
// ---------------------------------------------------------------------------
__global__ __launch_bounds__(32) void affinity_kernel(const float* __restrict__ x,
                                                      const float* __restrict__ cent,
                                                      float* __restrict__ dst,
                                                      int write_dense) {
    int gid  = blockIdx.x;              // NB * SHW * 16
    int row  = gid & 15;                // pixel row within block
    int t    = (gid >> 4) & (SHW - 1);  // block / superpixel id
    int b    = gid >> 12;
    int tr = t >> 4, tc = t & 15;
    int y  = tr * 16 + row;
    int x0 = tc * 16;

    int lane = threadIdx.x;             // wave32
    int half = lane >> 4;               // 0: K pair {4i,4i+1}; 1: {4i+2,4i+3}
    int m16  = lane & 15;               // M (pixel) for A, N (neighbor) for B

    // neighbor superpixel for B column N = m16 (sp clamped to 0 when invalid,
    // so the centroid-row pointer is always dereferenceable)
    int n  = m16;
    int dr = n / 3 - 1, dc = n % 3 - 1;
    int nr = tr + dr, nc = tc + dc;
    bool nvalid = (n < NC) && nr >= 0 && nr < SGRID && nc >= 0 && nc < SGRID;
    int  sp     = nvalid ? (nr * SGRID + nc) : 0;
    const float* crow = cent + ((size_t)b * SHW + sp) * CSTRIDE;

    const float* xb = x + (size_t)b * NC * HW;
    const int rowoff = y * WW + x0 + m16;
    v8f acc = {};

    #pragma unroll
    for (int i = 0; i < 3; ++i) {
        int k0 = 4 * i + 2 * half;      // channel index pair (k0, k0+1)
        int k1 = k0 + 1;
        // clamp padded channel indices to a valid plane; mask value afterwards
        int ka = (k0 < NC) ? k0 : 0;
        int kb = (k1 < NC) ? k1 : 0;
        float ax = xb[(size_t)ka * HW + rowoff];          // unconditional load
        float ay = xb[(size_t)kb * HW + rowoff];          // unconditional load
        float bx = crow[k0];   // rows zero-padded at 9..11 -> always safe
        float by = crow[k1];
        v2f a, bm;
        a.x  = (k0 < NC) ? ax : 0.f;    // v_cndmask, not exec branching
        a.y  = (k1 < NC) ? ay : 0.f;
        bm.x = nvalid ? bx : 0.f;
        bm.y = nvalid ? by : 0.f;
        acc = wmma_dot_16x16x4(a, bm, acc);
    }

    // spill 16x16 dot tile to LDS so each pixel's 9 scores land in one lane
    __shared__ float tile[256];
    #pragma unroll
    for (int j = 0; j < 8; ++j) {
        int M = j + 8 * half;           // C/D layout: VGPR j -> M = j (+8 for hi half)
        tile[M * 16 + m16] = acc[j];
    }
    __syncthreads();

    if (half == 0) {                    // lanes 0..15: one pixel each
        int m = m16;
        int p = y * WW + x0 + m;
        float score[NC];
        int   spn[NC];
        bool  val[NC];
        float mx = -INFINITY;
        #pragma unroll
        for (int q = 0; q < NC; ++q) {
            int qdr = q / 3 - 1, qdc = q % 3 - 1;
            int qr = tr + qdr, qc = tc + qdc;
            bool v = qr >= 0 && qr < SGRID && qc >= 0 && qc < SGRID;
            val[q] = v;
            spn[q] = v ? (qr * SGRID + qc) : 0;
            float dot = tile[m * 16 + q];
            float cn  = cent[((size_t)b * SHW + spn[q]) * CSTRIDE + 12]; // safe (sp=0)
            float sc  = v ? (2.0f * dot - cn) : -INFINITY;  // == -dist + |pf|^2 shift
            score[q] = sc;
            mx = fmaxf(mx, sc);
        }
        float e[NC];
        float sum = 0.f;
        #pragma unroll
        for (int q = 0; q < NC; ++q) {
            e[q] = val[q] ? __expf(score[q] - mx) : 0.f;
            sum += e[q];
        }
        float inv = 1.0f / sum;
        if (!write_dense) {
            #pragma unroll
            for (int q = 0; q < NC; ++q)
                dst[((size_t)b * NC + q) * HW + p] = e[q] * inv;
        } else {
            #pragma unroll
            for (int q = 0; q < NC; ++q)
                if (val[q])
                    dst[((size_t)b * SHW + spn[q]) * (size_t)HW + p] = e[q] * inv;
        }
    }
}

// ---------------------------------------------------------------------------
// Kernel 3: centroid update. One workgroup per (b, s); gathers the 3x3
// neighborhood of blocks (each thread owns one pixel slot per block), then
// deterministic LDS tree reduction of 10 accumulators (9 weighted channel
// sums + weight sum).
// ---------------------------------------------------------------------------
__global__ __launch_bounds__(256) void centroid_update_kernel(const float* __restrict__ x,
                                                              const float* __restrict__ aff,
                                                              float* __restrict__ cent) {
    int s = blockIdx.x & (SHW - 1);
    int b = blockIdx.x >> 8;
    int sr = s >> 4, sc = s & 15;
    int tid = threadIdx.x;
    int py = tid >> 4, px = tid & 15;

    const float* xb = x + (size_t)b * NC * HW;
    float wacc = 0.f;
    float facc[NC];
    #pragma unroll
    for (int c = 0; c < NC; ++c) facc[c] = 0.f;

    #pragma unroll
    for (int dn = 0; dn < 9; ++dn) {
        int dr = dn / 3 - 1, dc = dn % 3 - 1;
        int tr = sr + dr, tc = sc + dc;          // contributing block
        if (tr < 0 || tr >= SGRID || tc < 0 || tc >= SGRID) continue;
        // neighbor index k of pixels in block (tr,tc) that points at s:
        int k = (-dr + 1) * 3 + (-dc + 1);
        int y = tr * 16 + py, xc = tc * 16 + px;
        int p = y * WW + xc;
        float w = aff[((size_t)b * NC + k) * HW + p];
        wacc += w;
        #pragma unroll
        for (int c = 0; c < NC; ++c) facc[c] += w * xb[(size_t)c * HW + p];
    }

    __shared__ float red[256];
    __shared__ float out[NC + 1];
    #pragma unroll
    for (int q = 0; q < NC + 1; ++q) {
        red[tid] = (q < NC) ? facc[q] : wacc;
        __syncthreads();
        for (int off = 128; off > 0; off >>= 1) {
            if (tid < off) red[tid] += red[tid + off];
            __syncthreads();
        }
        if (tid == 0) out[q] = red[0];
        __syncthreads();
    }
    if (tid == 0) {
        float inv = 1.0f / (out[NC] + 1e-16f);
        float* row = cent + ((size_t)b * SHW + s) * CSTRIDE;
        float cn = 0.f;
        #pragma unroll
        for (int c = 0; c < NC; ++c) {
            float v = out[c] * inv;
            row[c] = v;
            cn += v * v;
        }
        row[9] = 0.f; row[10] = 0.f; row[11] = 0.f;
        row[12] = cn;
        row[13] = 0.f; row[14] = 0.f; row[15] = 0.f;
    }
}

// ---------------------------------------------------------------------------
// Host-side orchestration
// ---------------------------------------------------------------------------
extern "C" void kernel_launch(void* const* d_in, const int* in_sizes, int n_in,
                              void* d_out, int out_size, void* d_ws, size_t ws_size,
                              hipStream_t stream) {
    const float* x = (const float*)d_in[0];
    float* out = (float*)d_out;

    const size_t centFloats = (size_t)NB * SHW * CSTRIDE;       // 16384
    const size_t affFloats  = (size_t)NB * NC * HW;             // 2,359,296
    float* cent0 = (float*)d_ws;
    float* cent1 = cent0 + centFloats;
    // Affinity scratch: prefer workspace; otherwise borrow the (not yet
    // needed) 256 MB output buffer — it is zero-filled afterwards anyway.
    float* aff;
    if (ws_size >= (2 * centFloats + affFloats) * sizeof(float))
        aff = cent1 + centFloats;
    else
        aff = out;

    // 1) block-mean centroids
    centroid_mean_kernel<<<NB * SHW, 256, 0, stream>>>(x, cent0);
    // 2) iteration 0: affinities (WMMA) -> scratch
    affinity_kernel<<<NB * SHW * 16, 32, 0, stream>>>(x, cent0, aff, 0);
    // 3) weighted centroid update
    centroid_update_kernel<<<NB * SHW, 256, 0, stream>>>(x, aff, cent1);
    // 4) zero the dense output (bandwidth-dominant step: 256 MB)
    const long n4 = (long)NB * SHW * HW / 4;                    // 16,777,216 float4
    zero_kernel<<<4096, 256, 0, stream>>>((float4*)out, n4);
    // 5) iteration 1: affinities (WMMA) scattered into dense output
    affinity_kernel<<<NB * SHW * 16, 32, 0, stream>>>(x, cent1, out, 1);
}